// PointMatchingLoss_66614942760995
// MI455X (gfx1250) — compile-verified
//
#include <hip/hip_runtime.h>

// CDNA5 / gfx1250, wave32. WMMA f32 16x16x4 used for the batched
// target·pred dot-product term of the brute-force 1-NN distance.

typedef __attribute__((ext_vector_type(2))) float v2f;
typedef __attribute__((ext_vector_type(4))) float v4f;
typedef __attribute__((ext_vector_type(8))) float v8f;

#define BATCH 16
#define NPTS  4096
#define DIAM  100.0f   // MESH_DIAMETER[labels[b]] ; labels are all 0, table has 1 entry
#define LOSS_SCALE (1.0f / ((float)NPTS * DIAM * (float)BATCH))  // LOSS_WEIGHT=1

__global__ void pml_zero_out(float* out) { out[0] = 0.0f; }

__device__ __forceinline__ float vext8(const v8f& v, int r) {
    switch (r & 7) {
        case 0: return v.s0; case 1: return v.s1; case 2: return v.s2; case 3: return v.s3;
        case 4: return v.s4; case 5: return v.s5; case 6: return v.s6; default: return v.s7;
    }
}

__launch_bounds__(256)
__global__ void PointMatchingLoss_kernel(const float* __restrict__ pred_r,   // [B,3,3]
                                         const float* __restrict__ pred_t,   // [B,3]
                                         const float* __restrict__ gt_r,     // [B,3,3]
                                         const float* __restrict__ gt_t,     // [B,3]
                                         const float* __restrict__ pts,      // [B,N,3]
                                         const float* __restrict__ sf,       // [B]
                                         float* __restrict__ out)            // [1]
{
    // Transformed pred points, packed as {x, y, z, ||p||^2}. 4096 * 16B = 64KB LDS.
    __shared__ v4f lds_pred[NPTS];

    const int b    = blockIdx.x >> 5;   // 32 blocks per batch
    const int tg   = blockIdx.x & 31;   // tile group within batch
    const int tid  = threadIdx.x;
    const int lane = tid & 31;
    const int wave = tid >> 5;

    const float s = sf[b];
    // scale_xy and scale_depth (factor 1.0): all three components scaled by s
    const float spx = pred_t[b * 3 + 0] * s;
    const float spy = pred_t[b * 3 + 1] * s;
    const float spz = pred_t[b * 3 + 2] * s;
    const float sgx = gt_t[b * 3 + 0] * s;
    const float sgy = gt_t[b * 3 + 1] * s;
    const float sgz = gt_t[b * 3 + 2] * s;

    const float* __restrict__ PR = pred_r + b * 9;
    const float* __restrict__ GR = gt_r + b * 9;
    const float pr0 = PR[0], pr1 = PR[1], pr2 = PR[2];
    const float pr3 = PR[3], pr4 = PR[4], pr5 = PR[5];
    const float pr6 = PR[6], pr7 = PR[7], pr8 = PR[8];

    const float* __restrict__ P = pts + (size_t)b * NPTS * 3;

    // ---- Cooperative precompute: pred-transformed cloud -> LDS ----
    for (int i = tid; i < NPTS; i += 256) {
        const float x = P[i * 3 + 0];
        const float y = P[i * 3 + 1];
        const float z = P[i * 3 + 2];
        const float qx = fmaf(pr0, x, fmaf(pr1, y, fmaf(pr2, z, spx)));
        const float qy = fmaf(pr3, x, fmaf(pr4, y, fmaf(pr5, z, spy)));
        const float qz = fmaf(pr6, x, fmaf(pr7, y, fmaf(pr8, z, spz)));
        v4f q;
        q.x = qx; q.y = qy; q.z = qz;
        q.w = fmaf(qx, qx, fmaf(qy, qy, qz * qz));
        lds_pred[i] = q;
    }
    __syncthreads();

    // ---- This wave's 16 target rows (A-matrix 16x4, K = {x,y,z,0}) ----
    const int  tile = tg * 8 + wave;        // 0..255, uniform per wave
    const int  row  = lane & 15;            // M = row
    const int  ti   = tile * 16 + row;
    const bool hi   = lane >= 16;           // lanes 16..31 carry K=2,3

    const float x = P[ti * 3 + 0];
    const float y = P[ti * 3 + 1];
    const float z = P[ti * 3 + 2];
    const float tx = fmaf(GR[0], x, fmaf(GR[1], y, fmaf(GR[2], z, sgx)));
    const float ty = fmaf(GR[3], x, fmaf(GR[4], y, fmaf(GR[5], z, sgy)));
    const float tz = fmaf(GR[6], x, fmaf(GR[7], y, fmaf(GR[8], z, sgz)));
    const float tn2 = fmaf(tx, tx, fmaf(ty, ty, tz * tz));

    v2f a;
    a.x = hi ? tz : tx;   // VGPR0: K=0 (lanes 0-15) / K=2 (lanes 16-31)
    a.y = hi ? 0.0f : ty; // VGPR1: K=1               / K=3 (zero pad)

    v8f acc;
#pragma unroll
    for (int r = 0; r < 8; ++r) acc[r] = 3.0e38f;

    // ---- Sweep all 4096 pred points in 16-wide tiles ----
    for (int pt = 0; pt < NPTS / 16; ++pt) {
        const v4f q = lds_pred[pt * 16 + row];  // one ds_load_b128 / lane
        v2f bm;
        bm.x = hi ? q.z : q.x;  // B-matrix 4x16, mirrored K striping
        bm.y = hi ? 0.0f : q.y;
        v8f c = {};
        // D = A(16x4) x B(4x16), C = 0  ->  v_wmma_f32_16x16x4_f32
        v8f d = __builtin_amdgcn_wmma_f32_16x16x4_f32(
            /*neg_a=*/false, a, /*neg_b=*/false, bm,
            /*c_mod=*/(short)0, c, /*reuse_a=*/false, /*reuse_b=*/false);
        const float pn2 = q.w;
#pragma unroll
        for (int r = 0; r < 8; ++r) {
            const float cand = fmaf(-2.0f, d[r], pn2);  // ||p||^2 - 2 t.p
            acc[r] = fminf(acc[r], cand);
        }
    }

    // ---- Min over the 16 pred columns held by each 16-lane half ----
#pragma unroll
    for (int m = 8; m >= 1; m >>= 1) {
#pragma unroll
        for (int r = 0; r < 8; ++r)
            acc[r] = fminf(acc[r], __shfl_xor(acc[r], m, 32));
    }
    // Now: lanes 0-15 hold rows 0..7 in acc[0..7]; lanes 16-31 hold rows 8..15.

    // Row M finalized by: lane M (M<8) and lane M+16 (M>=8); tn2 there matches M.
    float contrib = 0.0f;
    if (lane < 8 || lane >= 24) {
        const float mn = vext8(acc, lane & 7);
        const float d2 = mn + tn2;
        contrib = sqrtf(fmaxf(d2, 0.0f)) * LOSS_SCALE;
    }
#pragma unroll
    for (int m = 16; m >= 1; m >>= 1) contrib += __shfl_xor(contrib, m, 32);
    if (lane == 0) atomicAdd(out, contrib);
}

extern "C" void kernel_launch(void* const* d_in, const int* in_sizes, int n_in,
                              void* d_out, int out_size, void* d_ws, size_t ws_size,
                              hipStream_t stream) {
    const float* pred_r = (const float*)d_in[0];
    const float* pred_t = (const float*)d_in[1];
    const float* gt_r   = (const float*)d_in[2];
    const float* gt_t   = (const float*)d_in[3];
    // d_in[4] = labels (int64, all zero; diameter table has a single entry -> 100.0)
    const float* pts    = (const float*)d_in[5];
    const float* sf     = (const float*)d_in[6];
    float* out = (float*)d_out;

    pml_zero_out<<<1, 1, 0, stream>>>(out);
    // 16 batches * 32 blocks, 256 threads (8 wave32 waves; 1 target tile per wave)
    PointMatchingLoss_kernel<<<dim3(BATCH * 32), dim3(256), 0, stream>>>(
        pred_r, pred_t, gt_r, gt_t, pts, sf, out);
    (void)in_sizes; (void)n_in; (void)out_size; (void)d_ws; (void)ws_size;
}